// SigmaMoELayer_1408749273685
// MI455X (gfx1250) — compile-verified
//
#include <hip/hip_runtime.h>
#include <math.h>

typedef float v2f __attribute__((ext_vector_type(2)));
typedef float v8f __attribute__((ext_vector_type(8)));

// Problem dims (fixed by reference)
#define B_  8
#define S_  2048
#define D_  1024
#define E_  32
#define ES_ 128
#define V_  1024
#define K_  4
#define T_  (B_*S_)      // 16384 tokens

// Tiling
#define KC        32
#define SA_STRIDE 34     // X tile   (64 x 32, padded)   -> conflict-free A-operand reads
#define SB_STRIDE 136    // B tile   (32 x 128, padded)  -> conflict-free B-operand reads
#define SH_STRIDE 132    // H buffer (64 x 128, padded)  -> conflict-free A-operand float2 reads

// ---------------- workspace layout ----------------
// [0,128)                      counts  int[32]
// [128,256)                    sums    float[32]   (per-expert sum of softmax probs)
// [256, 256+4*E*T)             tok_list  int[E][T]
// [+4*E*T]                     gate_list float[E][T]
// total ~4.2 MB

__global__ __launch_bounds__(256) void zero_init_kernel(float4* __restrict__ out4, int n4,
                                                        int* __restrict__ counts,
                                                        float* __restrict__ sums) {
  int i = blockIdx.x * 256 + threadIdx.x;
  if (i < n4) out4[i] = make_float4(0.f, 0.f, 0.f, 0.f);
  if (blockIdx.x == 0 && threadIdx.x < E_) { counts[threadIdx.x] = 0; sums[threadIdx.x] = 0.f; }
}

// One wave (32 lanes) per token; lane == expert. Computes router logits,
// reg-loss partial sums, top-4 selection, and per-expert token lists.
__global__ __launch_bounds__(256) void router_topk_kernel(
    const float* __restrict__ x, const float* __restrict__ Wsel,
    int* __restrict__ counts, float* __restrict__ sums,
    int* __restrict__ tok_list, float* __restrict__ gate_list) {
  extern __shared__ char smem[];
  float* Wl   = (float*)smem;                    // 256*33 floats (padded)
  float* esum = (float*)(smem + 256 * 33 * 4);   // 32 floats

  const int tid  = threadIdx.x;
  const int lane = tid & 31;
  const int w    = tid >> 5;
  const int tok  = blockIdx.x * 8 + w;

  if (tid < E_) esum[tid] = 0.f;

  float acc = 0.f;
  for (int dc = 0; dc < D_; dc += 256) {
    __syncthreads();
    // stage Wsel[e][dc..dc+256) as Wl[d_local*33 + e]; coalesced reads, padded writes
    #pragma unroll
    for (int e = 0; e < E_; ++e)
      Wl[tid * 33 + e] = Wsel[e * D_ + dc + tid];
    __syncthreads();
    const float* xp = x + (size_t)tok * D_ + dc;
    #pragma unroll 8
    for (int dl = 0; dl < 256; ++dl)
      acc = fmaf(xp[dl], Wl[dl * 33 + lane], acc);
  }

  // log-softmax stats across the 32 experts held in this wave's lanes
  float m = acc;
  #pragma unroll
  for (int off = 16; off > 0; off >>= 1) m = fmaxf(m, __shfl_xor(m, off));
  float p = expf(acc - m);
  float s = p;
  #pragma unroll
  for (int off = 16; off > 0; off >>= 1) s += __shfl_xor(s, off);
  float pe = p / s;                  // exp(logit - lse) in (0,1]
  atomicAdd(&esum[lane], pe);        // LDS f32 atomic

  // top-4 by logit (== top-4 by sigmoid); jax tie-break: lower index first
  float v = acc;
  #pragma unroll
  for (int k = 0; k < K_; ++k) {
    float bv = v; int bi = lane;
    #pragma unroll
    for (int off = 16; off > 0; off >>= 1) {
      float ov = __shfl_xor(bv, off);
      int   oi = __shfl_xor(bi, off);
      if (ov > bv || (ov == bv && oi < bi)) { bv = ov; bi = oi; }
    }
    if (lane == 0) {
      float g = 1.f / (1.f + expf(-bv));          // sigmoid gate
      int pos = atomicAdd(&counts[bi], 1);
      tok_list[bi * T_ + pos]  = tok;
      gate_list[bi * T_ + pos] = g;
    }
    if (lane == bi) v = -__builtin_inff();
  }

  __syncthreads();
  if (tid < E_) atomicAdd(&sums[tid], esum[tid]);
}

__global__ void reg_finalize_kernel(const float* __restrict__ sums, float* __restrict__ reg_out) {
  const int l = threadIdx.x;
  float q = sums[l] * (1.f / (float)T_);  // exp(lm[e])
  float val = q * logf(q);                // lm * exp(lm)
  #pragma unroll
  for (int off = 16; off > 0; off >>= 1) val += __shfl_xor(val, off);
  if (l == 0) reg_out[0] = val;
}

// Fused per-expert kernel: gather 64 token slots, GEMM1 (x @ keys[e]) with f32 WMMA,
// relu*gate into LDS, GEMM2 (h @ values[e]) with f32 WMMA, atomic scatter-add to out.
__global__ __launch_bounds__(256) void expert_kernel(
    const float* __restrict__ x, const float* __restrict__ keys,
    const float* __restrict__ values, const int* __restrict__ counts,
    const int* __restrict__ tok_list, const float* __restrict__ gate_list,
    float* __restrict__ out) {
  extern __shared__ char smem[];
  int*   smTok  = (int*)smem;                                       // 64
  float* smGate = (float*)(smem + 256);                             // 64
  float* smA    = (float*)(smem + 512);                             // 64*34
  float* smB    = (float*)(smem + 512 + 64 * SA_STRIDE * 4);        // 32*136
  float* smH    = (float*)(smem + 512 + 64 * SA_STRIDE * 4 + KC * SB_STRIDE * 4); // 64*132

  const int e    = blockIdx.x;
  const int cnt  = counts[e];
  const int base = blockIdx.y * 64;
  if (base >= cnt) return;                       // block-uniform exit
  const int nvalid = min(64, cnt - base);

  const int tid   = threadIdx.x;
  const int lane  = tid & 31;
  const int w     = tid >> 5;
  const int ln15  = lane & 15;
  const int khalf = (lane >> 4) << 1;            // K split {0,1}/{2,3} per ISA f32 A/B layout
  const int mrow  = (lane >> 4) * 8;             // C/D row split per ISA layout
  const int mtb   = (w & 1) * 2;                 // wave's 2 M-tiles
  const int ntb   = (w >> 1) * 2;                // wave's 2 N-tiles

  if (tid < 64) {
    if (tid < nvalid) {
      smTok[tid]  = tok_list[e * T_ + base + tid];
      smGate[tid] = gate_list[e * T_ + base + tid];
    } else { smTok[tid] = 0; smGate[tid] = 0.f; }
  }
  __syncthreads();

  const v8f vzero = {0.f,0.f,0.f,0.f,0.f,0.f,0.f,0.f};
  v8f acc[2][2];
  acc[0][0] = vzero; acc[0][1] = vzero; acc[1][0] = vzero; acc[1][1] = vzero;

  // staging coords
  const int am  = tid >> 2;           // X row 0..63
  const int akq = (tid & 3) * 8;      // X col base
  const int bkr = tid >> 3;           // B row 0..31
  const int bnb = (tid & 7) * 16;     // B col base
  const float* keyE = keys   + (size_t)e * (D_ * ES_);
  const float* valE = values + (size_t)e * (ES_ * V_);
  const int tokA = smTok[am];

  // ---------------- GEMM1: H(64x128) = relu(X(64x1024) @ keys(1024x128)) * gate ----
  for (int dc = 0; dc < D_; dc += KC) {
    { // stage gathered X tile
      const float* xp = x + (size_t)tokA * D_ + dc + akq;
      float4 u0 = *(const float4*)xp;
      float4 u1 = *(const float4*)(xp + 4);
      float* dp = smA + am * SA_STRIDE + akq;
      dp[0]=u0.x; dp[1]=u0.y; dp[2]=u0.z; dp[3]=u0.w;
      dp[4]=u1.x; dp[5]=u1.y; dp[6]=u1.z; dp[7]=u1.w;
    }
    { // stage keys tile (32 x 128)
      const float* kp = keyE + (size_t)(dc + bkr) * ES_ + bnb;
      float* dp = smB + bkr * SB_STRIDE + bnb;
      #pragma unroll
      for (int q = 0; q < 16; q += 4) {
        float4 u = *(const float4*)(kp + q);
        dp[q]=u.x; dp[q+1]=u.y; dp[q+2]=u.z; dp[q+3]=u.w;
      }
    }
    __syncthreads();
    #pragma unroll
    for (int k0 = 0; k0 < KC; k0 += 4) {
      v2f a0 = *(const v2f*)(smA + (mtb * 16 + ln15) * SA_STRIDE + k0 + khalf);
      v2f a1 = *(const v2f*)(smA + ((mtb + 1) * 16 + ln15) * SA_STRIDE + k0 + khalf);
      const int br = k0 + khalf;
      v2f b0, b1;
      b0.x = smB[br * SB_STRIDE + ntb * 16 + ln15];
      b0.y = smB[(br + 1) * SB_STRIDE + ntb * 16 + ln15];
      b1.x = smB[br * SB_STRIDE + (ntb + 1) * 16 + ln15];
      b1.y = smB[(br + 1) * SB_STRIDE + (ntb + 1) * 16 + ln15];
      acc[0][0] = __builtin_amdgcn_wmma_f32_16x16x4_f32(false, a0, false, b0, (short)0, acc[0][0], false, false);
      acc[0][1] = __builtin_amdgcn_wmma_f32_16x16x4_f32(false, a0, false, b1, (short)0, acc[0][1], false, false);
      acc[1][0] = __builtin_amdgcn_wmma_f32_16x16x4_f32(false, a1, false, b0, (short)0, acc[1][0], false, false);
      acc[1][1] = __builtin_amdgcn_wmma_f32_16x16x4_f32(false, a1, false, b1, (short)0, acc[1][1], false, false);
    }
    __syncthreads();
  }

  // epilogue1: relu * gate -> LDS H   (D layout: VGPR r -> row r + mrow, col = lane%16)
  #pragma unroll
  for (int mi = 0; mi < 2; ++mi)
    #pragma unroll
    for (int ni = 0; ni < 2; ++ni) {
      const int ncol = (ntb + ni) * 16 + ln15;
      #pragma unroll
      for (int r = 0; r < 8; ++r) {
        const int ml = (mtb + mi) * 16 + mrow + r;
        smH[ml * SH_STRIDE + ncol] = fmaxf(acc[mi][ni][r], 0.f) * smGate[ml];
      }
    }
  __syncthreads();

  // ---------------- GEMM2: out(64x1024) += H(64x128) @ values(128x1024) -------------
  for (int nc = 0; nc < V_; nc += 128) {
    acc[0][0] = vzero; acc[0][1] = vzero; acc[1][0] = vzero; acc[1][1] = vzero;
    for (int kc = 0; kc < ES_; kc += KC) {
      { // stage values tile (32 x 128)
        const float* vp = valE + (size_t)(kc + bkr) * V_ + nc + bnb;
        float* dp = smB + bkr * SB_STRIDE + bnb;
        #pragma unroll
        for (int q = 0; q < 16; q += 4) {
          float4 u = *(const float4*)(vp + q);
          dp[q]=u.x; dp[q+1]=u.y; dp[q+2]=u.z; dp[q+3]=u.w;
        }
      }
      __syncthreads();
      #pragma unroll
      for (int k0 = 0; k0 < KC; k0 += 4) {
        const int kk = kc + k0 + khalf;
        v2f a0 = *(const v2f*)(smH + (mtb * 16 + ln15) * SH_STRIDE + kk);
        v2f a1 = *(const v2f*)(smH + ((mtb + 1) * 16 + ln15) * SH_STRIDE + kk);
        const int br = k0 + khalf;
        v2f b0, b1;
        b0.x = smB[br * SB_STRIDE + ntb * 16 + ln15];
        b0.y = smB[(br + 1) * SB_STRIDE + ntb * 16 + ln15];
        b1.x = smB[br * SB_STRIDE + (ntb + 1) * 16 + ln15];
        b1.y = smB[(br + 1) * SB_STRIDE + (ntb + 1) * 16 + ln15];
        acc[0][0] = __builtin_amdgcn_wmma_f32_16x16x4_f32(false, a0, false, b0, (short)0, acc[0][0], false, false);
        acc[0][1] = __builtin_amdgcn_wmma_f32_16x16x4_f32(false, a0, false, b1, (short)0, acc[0][1], false, false);
        acc[1][0] = __builtin_amdgcn_wmma_f32_16x16x4_f32(false, a1, false, b0, (short)0, acc[1][0], false, false);
        acc[1][1] = __builtin_amdgcn_wmma_f32_16x16x4_f32(false, a1, false, b1, (short)0, acc[1][1], false, false);
      }
      __syncthreads();
    }
    // epilogue2: scatter-accumulate to out
    #pragma unroll
    for (int mi = 0; mi < 2; ++mi)
      #pragma unroll
      for (int ni = 0; ni < 2; ++ni) {
        const int ncol = nc + (ntb + ni) * 16 + ln15;
        #pragma unroll
        for (int r = 0; r < 8; ++r) {
          const int ml = (mtb + mi) * 16 + mrow + r;
          if (ml < nvalid)
            atomicAdd(out + (size_t)smTok[ml] * V_ + ncol, acc[mi][ni][r]);
        }
      }
  }
}

extern "C" void kernel_launch(void* const* d_in, const int* in_sizes, int n_in,
                              void* d_out, int out_size, void* d_ws, size_t ws_size,
                              hipStream_t stream) {
  const float* x      = (const float*)d_in[0];   // [B,S,D]
  const float* keys   = (const float*)d_in[1];   // [E,D,ES]
  const float* values = (const float*)d_in[2];   // [E,ES,V]
  const float* Wsel   = (const float*)d_in[3];   // [E,D]
  float* out = (float*)d_out;                    // [B,S,V] floats + 1 reg_loss float

  char* ws = (char*)d_ws;
  int*   counts    = (int*)ws;
  float* sums      = (float*)(ws + 128);
  int*   tok_list  = (int*)(ws + 256);
  float* gate_list = (float*)(ws + 256 + sizeof(int) * (size_t)E_ * T_);

  const int n4 = (T_ * V_) / 4;
  zero_init_kernel<<<(n4 + 255) / 256, 256, 0, stream>>>((float4*)out, n4, counts, sums);

  const size_t routerLds = (size_t)(256 * 33 * 4 + E_ * 4);
  router_topk_kernel<<<T_ / 8, 256, routerLds, stream>>>(x, Wsel, counts, sums, tok_list, gate_list);

  reg_finalize_kernel<<<1, 32, 0, stream>>>(sums, out + (size_t)T_ * V_);

  const size_t expLds = 512 + (size_t)64 * SA_STRIDE * 4 + (size_t)KC * SB_STRIDE * 4
                            + (size_t)64 * SH_STRIDE * 4;   // 60416 B
  dim3 grid(E_, T_ / 64);
  expert_kernel<<<grid, 256, expLds, stream>>>(x, keys, values, counts, tok_list, gate_list, out);
}